// SRT_AdaLISTA_31688268710427
// MI455X (gfx1250) — compile-verified
//
#include <hip/hip_runtime.h>
#include <hip/hip_bf16.h>

// ---------------------------------------------------------------------------
// AdaLISTA on MI455X (gfx1250):
//   - FFT/iFFT over the 512-bin band recast as banded DFT matmuls executed
//     with V_WMMA_F32_16X16X4_F32 (full f32 fidelity; compute is trivial vs
//     the bandwidth floor, so no precision downgrade needed).
//   - Per-bin complex matvec chain (forward + adjoint with diag(W1)/diag(W2))
//     is bandwidth-bound (0.25 FLOP/B, L-band fits in 192MB L2) -> VALU+LDS.
// ---------------------------------------------------------------------------

typedef __attribute__((ext_vector_type(2))) float v2f;
typedef __attribute__((ext_vector_type(8))) float v8f;

#define NFFT 2048
#define NB   512     // ihigh - ilow + 1
#define ILOW 1
#define NT   1024
#define NP   64
#define NQ   128

// workspace layout (float offsets)
#define OFF_TWF_COS  0                        // [NB, NT]  cos(2*pi*f*t/N)
#define OFF_TWF_NSIN (NB * NT)                // [NB, NT] -sin(2*pi*f*t/N)
#define OFF_TWI_COS  (2 * NB * NT)            // [NT, NB]  cos
#define OFF_TWI_NSIN (3 * NB * NT)            // [NT, NB] -sin
#define OFF_XF_RE    (4 * NB * NT)            // [NB, NQ]
#define OFF_XF_IM    (OFF_XF_RE + NB * NQ)
#define OFF_MB_RE    (OFF_XF_IM + NB * NQ)
#define OFF_MB_IM    (OFF_MB_RE + NB * NQ)
#define OFF_XCUR     (OFF_MB_IM + NB * NQ)    // [NT, NQ]
#define OFF_XNXT     (OFF_XCUR + NT * NQ)     // [NT, NQ]

// ---------------------------------------------------------------------------
// Twiddle tables; integer phase reduction keeps __sincosf args in [0, 2*pi).
__global__ void tw_init_kernel(float* __restrict__ ws) {
  int idx = blockIdx.x * blockDim.x + threadIdx.x;
  if (idx >= NB * NT) return;
  int fb = idx / NT;
  int t  = idx % NT;
  int f  = fb + ILOW;
  int ph = (f * t) & (NFFT - 1);
  float ang = (float)ph * (6.28318530717958647692f / (float)NFFT);
  float s, c;
  __sincosf(ang, &s, &c);
  ws[OFF_TWF_COS  + fb * NT + t] = c;
  ws[OFF_TWF_NSIN + fb * NT + t] = -s;
  ws[OFF_TWI_COS  + t * NB + fb] = c;
  ws[OFF_TWI_NSIN + t * NB + fb] = -s;
}

__global__ void xzero_kernel(float* __restrict__ x) {
  x[blockIdx.x * blockDim.x + threadIdx.x] = 0.0f;
}

// ---------------------------------------------------------------------------
// Forward banded DFT:  xf[512,128] = TW[512,1024] @ x[1024,128]  (re & im).
// One wave per 16x16 output tile; K swept 4 at a time by f32 WMMA.
__global__ __launch_bounds__(128) void fwd_dft_kernel(
    const float* __restrict__ tw_cos, const float* __restrict__ tw_nsin,
    const float* __restrict__ x,
    float* __restrict__ xf_re, float* __restrict__ xf_im) {
  const int wave = threadIdx.x >> 5;
  const int lane = threadIdx.x & 31;
  const int tile = blockIdx.x * 4 + wave;   // 0..255
  const int tm   = tile >> 3;               // bin tile   0..31
  const int tn   = tile & 7;                // q tile     0..7
  const int l15  = lane & 15;
  const int hi   = lane >> 4;               // K half: lanes 16-31 hold K+2,K+3
  const int row  = tm * 16 + l15;           // A row (bin)
  const int col  = tn * 16 + l15;           // B/C col (q)

  const float* __restrict__ Ac = tw_cos  + row * NT;
  const float* __restrict__ As = tw_nsin + row * NT;

  v8f acc_re = {};
  v8f acc_im = {};
#pragma unroll 4
  for (int kb = 0; kb < NT; kb += 4) {
    const int k0 = kb + hi * 2;
    v2f a_c = *(const v2f*)(Ac + k0);
    v2f a_s = *(const v2f*)(As + k0);
    v2f b;
    b.x = x[k0 * NQ + col];
    b.y = x[(k0 + 1) * NQ + col];
    acc_re = __builtin_amdgcn_wmma_f32_16x16x4_f32(false, a_c, false, b,
                                                   (short)0, acc_re, false, false);
    acc_im = __builtin_amdgcn_wmma_f32_16x16x4_f32(false, a_s, false, b,
                                                   (short)0, acc_im, false, false);
  }
#pragma unroll
  for (int v = 0; v < 8; ++v) {
    const int r = tm * 16 + v + hi * 8;     // C layout: rows v / v+8
    xf_re[r * NQ + col] = acc_re[v];
    xf_im[r * NQ + col] = acc_im[v];
  }
}

// ---------------------------------------------------------------------------
// Per-bin band operator (bandwidth-bound; one block per frequency bin):
//   yt[p] = d1[p] * sum_q L[f,p,q] * xf[q];  rb = yt - yb
//   mb[q] = d2[q] * sum_p conj(L[f,p,q]) * rb[p]
__global__ __launch_bounds__(128) void band_op_kernel(
    const float* __restrict__ L_re, const float* __restrict__ L_im,
    const float* __restrict__ y_re, const float* __restrict__ y_im,
    const float* __restrict__ W1,   const float* __restrict__ W2,
    const float* __restrict__ xf_re, const float* __restrict__ xf_im,
    float* __restrict__ mb_re, float* __restrict__ mb_im) {
  __shared__ float sxr[NQ], sxi[NQ], srr[NP], sri[NP];
  const int fb  = blockIdx.x;
  const int f   = fb + ILOW;
  const int tid = threadIdx.x;

  sxr[tid] = xf_re[fb * NQ + tid];
  sxi[tid] = xf_im[fb * NQ + tid];
  __syncthreads();

  const float* __restrict__ Lr = L_re + (size_t)f * NP * NQ;
  const float* __restrict__ Li = L_im + (size_t)f * NP * NQ;

  if (tid < NP) {
    const int p = tid;
    const float* lr = Lr + p * NQ;
    const float* li = Li + p * NQ;
    float ar = 0.0f, ai = 0.0f;
#pragma unroll 4
    for (int q = 0; q < NQ; ++q) {
      const float re = lr[q], im = li[q];
      ar += re * sxr[q] - im * sxi[q];
      ai += re * sxi[q] + im * sxr[q];
    }
    const float d1 = W1[p * NP + p];
    srr[p] = d1 * ar - y_re[f * NP + p];
    sri[p] = d1 * ai - y_im[f * NP + p];
  }
  __syncthreads();

  {
    const int q = tid;
    float ar = 0.0f, ai = 0.0f;
#pragma unroll 4
    for (int p = 0; p < NP; ++p) {
      const float re = Lr[p * NQ + q], im = Li[p * NQ + q];
      const float rr = srr[p], ri = sri[p];
      ar += re * rr + im * ri;     // conj(L) * rb
      ai += re * ri - im * rr;
    }
    const float d2 = W2[q * NQ + q];
    mb_re[fb * NQ + q] = d2 * ar;
    mb_im[fb * NQ + q] = d2 * ai;
  }
}

// ---------------------------------------------------------------------------
// Conjugate-symmetric inverse DFT (real output) fused with shrinkage:
//   x_tilde[t,q] = (2/NFFT) * sum_f [ cos*mb_re - sin*mb_im ]
//   z = x - gamma*x_tilde;  x' = sign(z)*max(|z|-theta, 0)
__global__ __launch_bounds__(128) void inv_dft_shrink_kernel(
    const float* __restrict__ twi_cos, const float* __restrict__ twi_nsin,
    const float* __restrict__ mb_re, const float* __restrict__ mb_im,
    const float* __restrict__ xcur,
    const float* __restrict__ gammas, const float* __restrict__ thetas,
    int layer, float* __restrict__ xout) {
  const int wave = threadIdx.x >> 5;
  const int lane = threadIdx.x & 31;
  const int tile = blockIdx.x * 4 + wave;   // 0..511
  const int tm   = tile >> 3;               // time tile 0..63
  const int tn   = tile & 7;                // q tile    0..7
  const int l15  = lane & 15;
  const int hi   = lane >> 4;
  const int row  = tm * 16 + l15;           // t
  const int col  = tn * 16 + l15;           // q

  const float* __restrict__ Ac = twi_cos  + row * NB;
  const float* __restrict__ As = twi_nsin + row * NB;

  v8f acc = {};
#pragma unroll 4
  for (int kb = 0; kb < NB; kb += 4) {
    const int k0 = kb + hi * 2;
    v2f a_c = *(const v2f*)(Ac + k0);
    v2f a_s = *(const v2f*)(As + k0);
    v2f b_r, b_i;
    b_r.x = mb_re[k0 * NQ + col];
    b_r.y = mb_re[(k0 + 1) * NQ + col];
    b_i.x = mb_im[k0 * NQ + col];
    b_i.y = mb_im[(k0 + 1) * NQ + col];
    acc = __builtin_amdgcn_wmma_f32_16x16x4_f32(false, a_c, false, b_r,
                                                (short)0, acc, false, false);
    acc = __builtin_amdgcn_wmma_f32_16x16x4_f32(false, a_s, false, b_i,
                                                (short)0, acc, false, false);
  }

  const float gamma = gammas[layer];
  const float theta = thetas[layer];
  const float scale = 2.0f / (float)NFFT;
#pragma unroll
  for (int v = 0; v < 8; ++v) {
    const int r = tm * 16 + v + hi * 8;
    const float xt = acc[v] * scale;
    const float z  = xcur[r * NQ + col] - gamma * xt;
    float az = fabsf(z) - theta;
    az = az > 0.0f ? az : 0.0f;
    xout[r * NQ + col] = copysignf(az, z);
  }
}

// ---------------------------------------------------------------------------
extern "C" void kernel_launch(void* const* d_in, const int* in_sizes, int n_in,
                              void* d_out, int out_size, void* d_ws, size_t ws_size,
                              hipStream_t stream) {
  // setup_inputs order: x0, y_re, y_im, L_re, L_im, W1, W2, gammas, thetas,
  //                     nt, ilow, ihigh  (x0 only supplies the output shape)
  const float* y_re   = (const float*)d_in[1];
  const float* y_im   = (const float*)d_in[2];
  const float* L_re   = (const float*)d_in[3];
  const float* L_im   = (const float*)d_in[4];
  const float* W1     = (const float*)d_in[5];
  const float* W2     = (const float*)d_in[6];
  const float* gammas = (const float*)d_in[7];
  const float* thetas = (const float*)d_in[8];
  const int n_layers  = in_sizes[7];

  float* ws = (float*)d_ws;
  float* TWF_C = ws + OFF_TWF_COS;
  float* TWF_S = ws + OFF_TWF_NSIN;
  float* TWI_C = ws + OFF_TWI_COS;
  float* TWI_S = ws + OFF_TWI_NSIN;
  float* XF_RE = ws + OFF_XF_RE;
  float* XF_IM = ws + OFF_XF_IM;
  float* MB_RE = ws + OFF_MB_RE;
  float* MB_IM = ws + OFF_MB_IM;
  float* xc    = ws + OFF_XCUR;
  float* xn    = ws + OFF_XNXT;

  tw_init_kernel<<<(NB * NT + 255) / 256, 256, 0, stream>>>(ws);
  xzero_kernel<<<(NT * NQ) / 256, 256, 0, stream>>>(xc);

  for (int i = 0; i < n_layers; ++i) {
    fwd_dft_kernel<<<64, 128, 0, stream>>>(TWF_C, TWF_S, xc, XF_RE, XF_IM);
    band_op_kernel<<<NB, 128, 0, stream>>>(L_re, L_im, y_re, y_im, W1, W2,
                                           XF_RE, XF_IM, MB_RE, MB_IM);
    float* dst = (i == n_layers - 1) ? (float*)d_out : xn;
    inv_dft_shrink_kernel<<<128, 128, 0, stream>>>(TWI_C, TWI_S, MB_RE, MB_IM,
                                                   xc, gammas, thetas, i, dst);
    float* t = xc; xc = xn; xn = t;
  }
}